// ExperimentalLoss_3616362463753
// MI455X (gfx1250) — compile-verified
//
#include <hip/hip_runtime.h>

typedef float v2f __attribute__((ext_vector_type(2)));
typedef float v4f __attribute__((ext_vector_type(4)));
typedef float v8f __attribute__((ext_vector_type(8)));

#define EPS_F 1e-7f
#define LN2_F 0.69314718056f
#define W160  (1.0f / (8400.0f * 160.0f))
#define W80   (1.0f / (8400.0f * 80.0f))

// lane' = ((lane & 0x1f) | 0) ^ MASK  -- group-of-32 ds_swizzle, no LDS memory.
// MASK must be an ICE -> template parameter.
template <int MASK>
__device__ __forceinline__ float xor_add(float s) {
    return s + __int_as_float(__builtin_amdgcn_ds_swizzle(
                   __float_as_int(s), (MASK << 10) | 0x1f));
}

// Full-wave (32 lane) fp32 sum using V_WMMA_F32_16X16X4_F32:
// A = ones(16x4), B = [x | 0] -> every D element is colsum[n] = x[n]+x[n+16];
// 4 xor-swizzles finish the 16-way sum. Requires EXEC all-ones.
__device__ __forceinline__ float wave_sum32(float x) {
    v2f a; a[0] = 1.0f; a[1] = 1.0f;
    v2f b; b[0] = x;    b[1] = 0.0f;
    v8f c = {};
    c = __builtin_amdgcn_wmma_f32_16x16x4_f32(
        /*neg_a=*/false, a, /*neg_b=*/false, b,
        /*c_mod=*/(short)0, c, /*reuse_a=*/false, /*reuse_b=*/false);
    float s = c[0];
    s = xor_add<1>(s);
    s = xor_add<2>(s);
    s = xor_add<4>(s);
    s = xor_add<8>(s);
    return s;                               // 32-lane sum in every lane
}

// 256-thread (8-wave) block sum. Valid in thread 0.
__device__ __forceinline__ float block_sum256(float x) {
    __shared__ float wsum[8];
    float s = wave_sum32(x);
    const int lane = threadIdx.x & 31;
    const int wid  = threadIdx.x >> 5;
    if (lane == 0) wsum[wid] = s;
    __syncthreads();
    float v = 0.0f;
    if (wid == 0) {                         // wave-uniform: EXEC stays full
        v = (lane < 8) ? wsum[lane] : 0.0f;
        v = xor_add<1>(v);
        v = xor_add<2>(v);
        v = xor_add<4>(v);
    }
    return v;
}

// Scaled focal term: t*q^g*log2(p) + (1-t)*p^g*log2(1-p)   (<= 0).
// The caller folds (-ln2 * mean-weight) into one per-thread constant.
// seg: 0 -> gamma 2.5 (x^2*sqrt x), 1 -> 2.0 (x^2), 2 -> 3.0 (x^3).
__device__ __forceinline__ float focal2(float p, float t, int seg) {
    p = fminf(fmaxf(p, EPS_F), 1.0f - EPS_F);
    const float q  = 1.0f - p;
    const float lp = __builtin_amdgcn_logf(p);    // v_log_f32 (log2)
    const float lq = __builtin_amdgcn_logf(q);
    const float sp = __builtin_amdgcn_sqrtf(p);   // raw v_sqrt_f32
    const float sq = __builtin_amdgcn_sqrtf(q);
    const float mp = (seg == 0) ? sp : ((seg == 2) ? p : 1.0f);
    const float mq = (seg == 0) ? sq : ((seg == 2) ? q : 1.0f);
    const float pg = p * p * mp;
    const float qg = q * q * mq;
    return fmaf(t, qg * lp, (1.0f - t) * pg * lq);
}

// One "chunk" = 8 consecutive floats (two B128 per tensor). A 720-float row is
// 90 chunks; gamma boundaries (320, 640, 720) are multiples of 8, so a chunk
// never straddles. gridDim*blockDim is kept a multiple of 90, so the
// chunk-in-row phase j (hence seg and weight) is loop-invariant per thread.
__global__ void __launch_bounds__(256)
focal_partial(const float* __restrict__ pred,
              const float* __restrict__ targ,
              float* __restrict__ partials,
              int n8) {
    const v4f* __restrict__ P = reinterpret_cast<const v4f*>(pred);
    const v4f* __restrict__ T = reinterpret_cast<const v4f*>(targ);
    const int stride = gridDim.x * blockDim.x;          // == 0 mod 90
    const int c0     = blockIdx.x * blockDim.x + threadIdx.x;
    const int j      = c0 % 90;                         // hoisted urem
    const int seg    = (j < 40) ? 0 : ((j < 80) ? 1 : 2);
    const float w    = ((j < 80) ? W160 : W80) * (-LN2_F);
    float acc = 0.0f;
    for (int c = c0; c < n8; c += stride) {
        const v4f pa = __builtin_nontemporal_load(&P[2 * c]);     // b128 NT
        const v4f pb = __builtin_nontemporal_load(&P[2 * c + 1]);
        const v4f ta = __builtin_nontemporal_load(&T[2 * c]);
        const v4f tb = __builtin_nontemporal_load(&T[2 * c + 1]);
        float s = focal2(pa[0], ta[0], seg) + focal2(pa[1], ta[1], seg)
                + focal2(pa[2], ta[2], seg) + focal2(pa[3], ta[3], seg)
                + focal2(pb[0], tb[0], seg) + focal2(pb[1], tb[1], seg)
                + focal2(pb[2], tb[2], seg) + focal2(pb[3], tb[3], seg);
        acc += s;
    }
    const float total = block_sum256(acc * w);          // all threads reach here
    if (threadIdx.x == 0) partials[blockIdx.x] = total;
}

__global__ void __launch_bounds__(256)
focal_finalize(const float* __restrict__ partials, int n,
               float* __restrict__ out) {
    float acc = 0.0f;
    for (int i = threadIdx.x; i < n; i += blockDim.x) acc += partials[i];
    const float total = block_sum256(acc);
    if (threadIdx.x == 0) out[0] = total;
}

extern "C" void kernel_launch(void* const* d_in, const int* in_sizes, int n_in,
                              void* d_out, int out_size, void* d_ws, size_t ws_size,
                              hipStream_t stream) {
    const float* pred = (const float*)d_in[0];
    const float* targ = (const float*)d_in[1];
    float* out        = (float*)d_out;
    float* partials   = (float*)d_ws;

    const int n  = in_sizes[0];            // 32 * 8400 * 720 = 193,536,000
    const int n8 = n / 8;                  // 24,192,000 chunks

    // blocks must be a multiple of 45 so stride = blocks*256 == 0 mod 90.
    int blocks = 4050;                     // 45 * 90; 16.2 KB of partials
    const int maxPartials = (int)(ws_size / sizeof(float));
    if (blocks > maxPartials) {
        blocks = (maxPartials / 45) * 45;
        if (blocks < 45) blocks = 45;      // assume ws_size >= 180 B
    }

    focal_partial<<<blocks, 256, 0, stream>>>(pred, targ, partials, n8);
    focal_finalize<<<1, 256, 0, stream>>>(partials, blocks, out);
}